// GraphAttentionLayer_34806414967136
// MI455X (gfx1250) — compile-verified
//
#include <hip/hip_runtime.h>
#include <math.h>

// Problem constants (from reference)
#define NN    50000
#define EE    800000
#define INF_  256
#define OUTF  64
#define NHEAD 4
#define ALPHA_ 0.2f
#define HDIM  (NHEAD * OUTF)   // 256 = concatenated head width

typedef __attribute__((ext_vector_type(2))) float v2f;
typedef __attribute__((ext_vector_type(8))) float v8f;

// ---------------------------------------------------------------------------
// Kernel 1: h = x @ W_cat   (N x 256) @ (256 x 256) -> (N x 256)
// W is (HEADS, IN_F, OUT_F); column c of W_cat is W[c/64][k][c%64].
// One wave32 per 16x16 output tile, fp32 WMMA 16x16x4, K-loop of 64 steps.
// VGPR layouts per CDNA5 ISA 7.12.2:
//   A (16x4):  lanes 0-15 -> M=lane, {K=k0,k0+1}; lanes 16-31 -> {K=k0+2,k0+3}
//   B (4x16):  lanes 0-15 -> N=lane, {K=k0,k0+1}; lanes 16-31 -> {K=k0+2,k0+3}
//   C (16x16): VGPR v -> row v + 8*(lane/16), col = lane%16
// ---------------------------------------------------------------------------
__global__ __launch_bounds__(256) void k_proj_gemm(const float* __restrict__ X,
                                                   const float* __restrict__ Wf,
                                                   float* __restrict__ H) {
  const int wave = threadIdx.x >> 5;
  const int lane = threadIdx.x & 31;
  const int half = lane >> 4;
  const int lr   = lane & 15;
  const int tile = blockIdx.x * 8 + wave;           // 3125 row-tiles * 16 col-tiles
  if (tile >= (NN / 16) * (HDIM / 16)) return;      // uniform per wave
  const int row0 = (tile >> 4) * 16;
  const int col0 = (tile & 15) * 16;

  const int ncol = col0 + lr;                       // output column (head*64+o)
  const float* __restrict__ Bcol = Wf + (size_t)(ncol >> 6) * (INF_ * OUTF) + (ncol & 63);
  const float* __restrict__ Arow = X + (size_t)(row0 + lr) * INF_;

  v8f c = {};
  #pragma unroll 8
  for (int k0 = 0; k0 < INF_; k0 += 4) {
    const int ka = k0 + (half << 1);
    v2f a = *(const v2f*)(Arow + ka);               // contiguous 8B load
    v2f b;
    b.x = Bcol[(size_t)ka * OUTF];                  // coalesced across lr
    b.y = Bcol[(size_t)(ka + 1) * OUTF];
    c = __builtin_amdgcn_wmma_f32_16x16x4_f32(false, a, false, b, (short)0, c, false, false);
  }
  float* __restrict__ Cp = H + ncol;
  #pragma unroll
  for (int v = 0; v < 8; ++v) {
    const int m = row0 + v + (half << 3);
    Cp[(size_t)m * HDIM] = c[v];
  }
}

// ---------------------------------------------------------------------------
// Kernel 2: s_src[h][n] = dot(h[h,n,:], a[h,0:64]);  s_tgt uses a[h,64:128]
// ---------------------------------------------------------------------------
__global__ __launch_bounds__(256) void k_scores(const float* __restrict__ H,
                                                const float* __restrict__ A,
                                                float* __restrict__ s_src,
                                                float* __restrict__ s_tgt) {
  const int idx = blockIdx.x * blockDim.x + threadIdx.x;
  if (idx >= NHEAD * NN) return;
  const int h = idx / NN;
  const int n = idx - h * NN;
  const float* __restrict__ hp = H + (size_t)n * HDIM + h * OUTF;
  const float* __restrict__ as = A + h * (2 * OUTF);
  const float* __restrict__ at = as + OUTF;
  float ss = 0.f, st = 0.f;
  #pragma unroll 8
  for (int o = 0; o < OUTF; ++o) {
    const float v = hp[o];
    ss += v * as[o];
    st += v * at[o];
  }
  s_src[idx] = ss;
  s_tgt[idx] = st;
}

// ---------------------------------------------------------------------------
// Init: zero h_prime + deg, set segmax = -inf (must run every launch)
// ---------------------------------------------------------------------------
__global__ __launch_bounds__(256) void k_init(float* __restrict__ Hp,
                                              float* __restrict__ segmax,
                                              float* __restrict__ deg) {
  const size_t i = (size_t)blockIdx.x * blockDim.x + threadIdx.x;
  if (i < (size_t)NN * HDIM) Hp[i] = 0.f;
  if (i < (size_t)NHEAD * NN) segmax[i] = -__builtin_inff();
  if (i < (size_t)NN) deg[i] = 0.f;
}

// ---------------------------------------------------------------------------
// Edge pass 1: e = leaky_relu(s_src[src]+s_tgt[tgt]); segmax/deg accumulation
// ---------------------------------------------------------------------------
__global__ __launch_bounds__(256) void k_edge1(const int* __restrict__ src,
                                               const int* __restrict__ tgt,
                                               const float* __restrict__ s_src,
                                               const float* __restrict__ s_tgt,
                                               float* __restrict__ ebuf,
                                               float* __restrict__ segmax,
                                               float* __restrict__ deg) {
  const int e = blockIdx.x * blockDim.x + threadIdx.x;
  if (e >= EE) return;
  const int s = src[e];
  const int t = tgt[e];
  #pragma unroll
  for (int h = 0; h < NHEAD; ++h) {
    float v = s_src[h * NN + s] + s_tgt[h * NN + t];
    v = v > 0.f ? v : ALPHA_ * v;
    ebuf[(size_t)e * NHEAD + h] = v;
    atomicMax(&segmax[h * NN + s], v);              // global_atomic_max_num_f32
  }
  atomicAdd(&deg[s], 1.0f);
}

// ---------------------------------------------------------------------------
// Node pass: rowmax = max(segmax,0) (in place); denom base = (N-deg)*exp(-rowmax)
// ---------------------------------------------------------------------------
__global__ __launch_bounds__(256) void k_node(float* __restrict__ segmax,
                                              const float* __restrict__ deg,
                                              float* __restrict__ denom) {
  const int idx = blockIdx.x * blockDim.x + threadIdx.x;
  if (idx >= NHEAD * NN) return;
  const int h = idx / NN;
  const int n = idx - h * NN;
  const float m = fmaxf(segmax[idx], 0.f);          // -inf (no edges) -> 0
  segmax[idx] = m;                                  // reuse as rowmax
  denom[idx] = ((float)NN - deg[n]) * __expf(-m);
}

// ---------------------------------------------------------------------------
// Edge pass 2: exp_e = exp(e - rowmax[src]) (in place); denom += exp_e
// ---------------------------------------------------------------------------
__global__ __launch_bounds__(256) void k_edge2(const int* __restrict__ src,
                                               float* __restrict__ ebuf,
                                               const float* __restrict__ rowmax,
                                               float* __restrict__ denom) {
  const int e = blockIdx.x * blockDim.x + threadIdx.x;
  if (e >= EE) return;
  const int s = src[e];
  #pragma unroll
  for (int h = 0; h < NHEAD; ++h) {
    const float ee = __expf(ebuf[(size_t)e * NHEAD + h] - rowmax[h * NN + s]);
    ebuf[(size_t)e * NHEAD + h] = ee;
    atomicAdd(&denom[h * NN + s], ee);
  }
}

// ---------------------------------------------------------------------------
// Edge pass 3 (scatter): h_prime[src] += h[tgt] * attn.  One block per edge,
// thread = (head, out) so the 64-wide h[tgt] gather and atomic scatter are
// contiguous.
// ---------------------------------------------------------------------------
__global__ __launch_bounds__(256) void k_edge3(const int* __restrict__ src,
                                               const int* __restrict__ tgt,
                                               const float* __restrict__ ebuf,
                                               const float* __restrict__ denom,
                                               const float* __restrict__ H,
                                               float* __restrict__ Hp) {
  const int e = blockIdx.x;
  const int s = src[e];
  const int t = tgt[e];
  __shared__ float sattn[NHEAD];
  if (threadIdx.x < NHEAD)
    sattn[threadIdx.x] = ebuf[(size_t)e * NHEAD + threadIdx.x] /
                         denom[threadIdx.x * NN + s];
  __syncthreads();
  const int h = threadIdx.x >> 6;
  const int o = threadIdx.x & 63;
  const float val = H[(size_t)t * HDIM + h * OUTF + o] * sattn[h];
  atomicAdd(&Hp[(size_t)s * HDIM + h * OUTF + o], val);
}

// ---------------------------------------------------------------------------
// Kernel 6: out = h_prime @ out_proj_w^T + b    (N x 256)@(256 x 64) -> (N x 64)
// Same fp32 WMMA tiling; B(k,j) = P[j*256 + k] (P is 64 KB -> L2 resident).
// ---------------------------------------------------------------------------
__global__ __launch_bounds__(256) void k_out_gemm(const float* __restrict__ Hp,
                                                  const float* __restrict__ P,
                                                  const float* __restrict__ bias,
                                                  float* __restrict__ Out) {
  const int wave = threadIdx.x >> 5;
  const int lane = threadIdx.x & 31;
  const int half = lane >> 4;
  const int lr   = lane & 15;
  const int tile = blockIdx.x * 8 + wave;           // 3125 * 4 tiles
  if (tile >= (NN / 16) * (OUTF / 16)) return;
  const int row0 = (tile >> 2) * 16;
  const int col0 = (tile & 3) * 16;

  const int ncol = col0 + lr;
  const float* __restrict__ Brow = P + (size_t)ncol * HDIM;   // row j of out_proj_w
  const float* __restrict__ Arow = Hp + (size_t)(row0 + lr) * HDIM;

  v8f c = {};
  #pragma unroll 8
  for (int k0 = 0; k0 < HDIM; k0 += 4) {
    const int ka = k0 + (half << 1);
    v2f a = *(const v2f*)(Arow + ka);
    v2f b = *(const v2f*)(Brow + ka);
    c = __builtin_amdgcn_wmma_f32_16x16x4_f32(false, a, false, b, (short)0, c, false, false);
  }
  const float bb = bias[ncol];
  float* __restrict__ Cp = Out + ncol;
  #pragma unroll
  for (int v = 0; v < 8; ++v) {
    const int m = row0 + v + (half << 3);
    Cp[(size_t)m * OUTF] = c[v] + bb;
  }
}

// ---------------------------------------------------------------------------
extern "C" void kernel_launch(void* const* d_in, const int* in_sizes, int n_in,
                              void* d_out, int out_size, void* d_ws, size_t ws_size,
                              hipStream_t stream) {
  const float* x    = (const float*)d_in[0];
  const int*   ei   = (const int*)d_in[1];        // (2, E): src row then tgt row
  const float* W    = (const float*)d_in[2];      // (4, 256, 64)
  const float* a    = (const float*)d_in[3];      // (4, 128, 1)
  const float* opw  = (const float*)d_in[4];      // (64, 256)
  const float* opb  = (const float*)d_in[5];      // (64,)
  float*       out  = (float*)d_out;              // (N, 64)

  const int* src = ei;
  const int* tgt = ei + EE;

  // Workspace carve-up (floats)
  float* ws     = (float*)d_ws;
  float* H      = ws;                              size_t off = (size_t)NN * HDIM;   // 12.8M
  float* Hp     = ws + off;                        off += (size_t)NN * HDIM;         // 12.8M
  float* s_src  = ws + off;                        off += (size_t)NHEAD * NN;
  float* s_tgt  = ws + off;                        off += (size_t)NHEAD * NN;
  float* segmax = ws + off;                        off += (size_t)NHEAD * NN;        // reused as rowmax
  float* denom  = ws + off;                        off += (size_t)NHEAD * NN;
  float* deg    = ws + off;                        off += (size_t)NN;
  float* ebuf   = ws + off;                        off += (size_t)EE * NHEAD;        // reused for exp_e

  // 0) init accumulators (largest span: N*HDIM)
  {
    const size_t n = (size_t)NN * HDIM;
    k_init<<<(unsigned)((n + 255) / 256), 256, 0, stream>>>(Hp, segmax, deg);
  }
  // 1) projection GEMM (WMMA fp32)
  k_proj_gemm<<<(NN / 16) * (HDIM / 16) / 8, 256, 0, stream>>>(x, W, H);
  // 2) per-node attention scores
  k_scores<<<(NHEAD * NN + 255) / 256, 256, 0, stream>>>(H, a, s_src, s_tgt);
  // 3) edge scores + segment max + degree
  k_edge1<<<(EE + 255) / 256, 256, 0, stream>>>(src, tgt, s_src, s_tgt, ebuf, segmax, deg);
  // 4) rowmax + denom base
  k_node<<<(NHEAD * NN + 255) / 256, 256, 0, stream>>>(segmax, deg, denom);
  // 5) exp + segment sum
  k_edge2<<<(EE + 255) / 256, 256, 0, stream>>>(src, ebuf, segmax, denom);
  // 6) weighted scatter into h_prime
  k_edge3<<<EE, 256, 0, stream>>>(src, tgt, ebuf, denom, H, Hp);
  // 7) output projection GEMM (WMMA fp32) + bias
  k_out_gemm<<<((NN / 16) * (OUTF / 16) + 7) / 8, 256, 0, stream>>>(Hp, opw, opb, out);
}